// _DepthwiseFIRConv1d_31877247271427
// MI455X (gfx1250) — compile-verified
//
#include <hip/hip_runtime.h>
#include <hip/hip_bf16.h>
#include <stdint.h>

// Problem constants (match reference setup_inputs)
#define NB    8
#define LSEQ  4096
#define NC    2048        // h*d = 16*128
#define KF    31
#define TTILE 512         // timesteps per block
#define TPB   256
#define CBLK  (TPB * 2)   // 512 channels per block (2 per thread as float2)
#define WELEM (CBLK * KF) // 15872 f32 filter elements staged in LDS (62 KB)

typedef unsigned int v4u __attribute__((ext_vector_type(4)));
typedef int          v4i __attribute__((ext_vector_type(4)));
typedef int          v8i __attribute__((ext_vector_type(8)));
typedef float        v2f __attribute__((ext_vector_type(2)));

__global__ __launch_bounds__(TPB) void fir_dwconv_kernel(
    const float* __restrict__ x, const float* __restrict__ w, float* __restrict__ y)
{
  __shared__ float wlds[WELEM];

  const int c0 = blockIdx.x * CBLK;
  const int t0 = blockIdx.y * TTILE;   // multiple of 32
  const int b  = blockIdx.z;

  // ---- Stage this block's 512 filters into LDS with the Tensor Data Mover.
  // 1-D descriptor: tile_dim0 = 15872 f32 contiguous (filters are (C,31), k
  // innermost, so a channel range is flat-contiguous). Wave 0 issues the DMA,
  // waits TENSORcnt, then the workgroup barrier publishes LDS to all waves.
  if (threadIdx.x < 32) {
    const uint32_t lds_off = (uint32_t)(size_t)(&wlds[0]);
    const uint64_t gaddr   = (uint64_t)(size_t)(w + (size_t)c0 * KF);

    v4u g0;
    g0[0] = 1u;                        // count=1 (valid), no gather/restore
    g0[1] = lds_off;                   // lds_addr (bytes)
    g0[2] = (uint32_t)gaddr;           // global_addr[31:0]
    g0[3] = ((uint32_t)(gaddr >> 32) & 0x01FFFFFFu) | 0x80000000u; // addr[56:32] | type=2

    v8i g1;
    g1[0] = 0x00020000;                              // workgroup_mask=0, data_size=4B
    g1[1] = (int)((uint32_t)WELEM << 16);            // tensor_dim0 lo16
    g1[2] = (int)(((uint32_t)WELEM >> 16) | (1u << 16)); // dim0 hi16 | tensor_dim1=1
    g1[3] = (int)((uint32_t)WELEM << 16);            // tile_dim0 = WELEM
    g1[4] = 1;                                       // tile_dim1 = 1, tile_dim2 = 0
    g1[5] = (int)WELEM;                              // tensor_dim0_stride lo32
    g1[6] = 0;
    g1[7] = 0;

    v4i gz = {0, 0, 0, 0};
#if __clang_major__ >= 23
    v8i gz8 = {0, 0, 0, 0, 0, 0, 0, 0};
    __builtin_amdgcn_tensor_load_to_lds(g0, g1, gz, gz, gz8, 0);
#else
    __builtin_amdgcn_tensor_load_to_lds(g0, g1, gz, gz, 0);
#endif
    __builtin_amdgcn_s_wait_tensorcnt(0);
  }
  __syncthreads();

  // ---- Per-thread filter pair (channels c, c+1) from LDS into registers
  // (lowers to ds_load_b128 runs; one-time cost).
  const int cl = threadIdx.x * 2;
  float2 wv[KF];
#pragma unroll
  for (int k = 0; k < KF; ++k) {
    wv[k].x = wlds[cl * KF + k];
    wv[k].y = wlds[(cl + 1) * KF + k];
  }

  const int c = c0 + cl;
  const float2* __restrict__ xb = (const float2*)(x + (size_t)b * LSEQ * NC + c);
  float*                     yp = y + (size_t)b * LSEQ * NC + c;
  const int rs = NC / 2;  // float2 stride per timestep (1024)

  // ---- Register ring buffer, period-32 so all indices are static after
  // unrolling (zero v_mov window shifting). Slot for time t is (t & 31);
  // t0 % 32 == 0 so the 30-tap lookback primes slots 2..31.
  float2 ring[32];
#pragma unroll
  for (int i = 0; i < 30; ++i) {
    const int tp = t0 - 30 + i;
    float2 v; v.x = 0.f; v.y = 0.f;
    if (tp >= 0) v = xb[(size_t)tp * rs];   // block-uniform predicate
    ring[i + 2] = v;
  }
  ring[0].x = 0.f; ring[0].y = 0.f;
  ring[1].x = 0.f; ring[1].y = 0.f;

  for (int it = 0; it < TTILE / 32; ++it) {
    const int tbase = t0 + it * 32;
    if (tbase + 64 < LSEQ) {
      // stream-prefetch two panels ahead (global_prefetch_b8)
      __builtin_prefetch(&xb[(size_t)(tbase + 64) * rs], 0, 1);
    }
#pragma unroll
    for (int j = 0; j < 32; ++j) {
      const size_t t = (size_t)(tbase + j);
      ring[j] = xb[t * rs];                 // newest sample -> slot j
      float2 a0, a1;
      a0.x = 0.f; a0.y = 0.f; a1.x = 0.f; a1.y = 0.f;
#pragma unroll
      for (int k = 0; k < KF; k += 2) {
        const float2 v0 = ring[(j + 2 + k) & 31];
        a0.x = fmaf(wv[k].x, v0.x, a0.x);
        a0.y = fmaf(wv[k].y, v0.y, a0.y);
        if (k + 1 < KF) {
          const float2 v1 = ring[(j + 3 + k) & 31];
          a1.x = fmaf(wv[k + 1].x, v1.x, a1.x);
          a1.y = fmaf(wv[k + 1].y, v1.y, a1.y);
        }
      }
      v2f acc;
      acc.x = a0.x + a1.x;
      acc.y = a0.y + a1.y;
      // y is write-once / never re-read: non-temporal store keeps L2 free
      // for the inbound x stream the prefetches are staging.
      __builtin_nontemporal_store(acc, (v2f*)(yp + t * NC));
    }
  }
}

extern "C" void kernel_launch(void* const* d_in, const int* in_sizes, int n_in,
                              void* d_out, int out_size, void* d_ws, size_t ws_size,
                              hipStream_t stream) {
  (void)in_sizes; (void)n_in; (void)out_size; (void)d_ws; (void)ws_size;
  const float* x = (const float*)d_in[0];
  const float* w = (const float*)d_in[1];
  float* y = (float*)d_out;
  dim3 grid(NC / CBLK, LSEQ / TTILE, NB);
  fir_dwconv_kernel<<<grid, TPB, 0, stream>>>(x, w, y);
}